// BoundaryUnit_10307921511134
// MI455X (gfx1250) — compile-verified
//
#include <hip/hip_runtime.h>
#include <hip/hip_bf16.h>
#include <math.h>

typedef __attribute__((ext_vector_type(2))) float v2f;
typedef __attribute__((ext_vector_type(4))) float f4;
typedef __attribute__((ext_vector_type(8))) float v8f;

#define BB 8
#define NN 128
#define LL 30
#define LP 32
#define DD 256
#define SCALE 0.0625f           // 1/sqrt(256)
#define NEGV  -1000000000.0f
#define LOG2E 1.4426950408889634f

__device__ __forceinline__ float fast_exp(float x) {
    return __builtin_amdgcn_exp2f(x * LOG2E);
}
__device__ __forceinline__ float fast_sigmoid(float x) {
    return __builtin_amdgcn_rcpf(1.0f + __builtin_amdgcn_exp2f(-LOG2E * x));
}

// ---------------------------------------------------------------------------
// Kernel 1: q = f_b @ Wq^T + bq   (B,N,D) ;  k = f_w @ Wk^T + bk  (B,LP,D)
// One wave (32 threads) per 16x16 output tile, V_WMMA_F32_16X16X4_F32.
// A layout (16x4 f32): lanes 0-15 -> M=l16, K=k0+{0,1}; lanes 16-31 -> K=k0+{2,3}
// B layout (4x16 f32): mirror, N=l16 per lane.
// C layout: VGPR r: lanes0-15 M=r, lanes16-31 M=8+r, N=l16.
// Stores are always in-range by construction (k padded to LP=32 rows, with
// zero A-mask for rows >= 30), so no exec-mask branching on the store path.
// ---------------------------------------------------------------------------
__global__ __launch_bounds__(32) void qk_kernel(
    const float* __restrict__ f_b, const float* __restrict__ f_w,
    const float* __restrict__ Wq,  const float* __restrict__ bq,
    const float* __restrict__ Wk,  const float* __restrict__ bk,
    float* __restrict__ qout, float* __restrict__ kout)
{
    int blk  = blockIdx.x;
    int lane = threadIdx.x;
    int half = lane >> 4;
    int l16  = lane & 15;

    const float *src, *W, *bias;
    float* dst;
    int rtile, dtile, srcRows;
    if (blk < 1024) {                  // q path: 8 b * 8 ntile * 16 dtile
        int b = blk >> 7; int rem = blk & 127;
        rtile = rem >> 4; dtile = rem & 15;
        src = f_b + b * NN * DD; W = Wq; bias = bq;
        dst = qout + b * NN * DD; srcRows = NN;
    } else {                           // k path: 8 b * 2 rtile * 16 dtile
        int blk2 = blk - 1024;
        int b = blk2 >> 5; int rem = blk2 & 31;
        rtile = rem >> 4; dtile = rem & 15;
        src = f_w + b * LL * DD; W = Wk; bias = bk;
        dst = kout + b * LP * DD; srcRows = LL;
    }

    int row   = rtile * 16 + l16;
    int rowc  = row < srcRows ? row : (srcRows - 1);
    float amask = row < srcRows ? 1.0f : 0.0f;
    int dcol  = dtile * 16 + l16;

    const float* aptr = src + rowc * DD + 2 * half;
    const float* bptr = W + dcol * DD + 2 * half;

    v8f c = {};
    #pragma unroll 4
    for (int k0 = 0; k0 < DD; k0 += 4) {
        v2f a = *(const v2f*)(aptr + k0);
        a *= amask;
        v2f bb = *(const v2f*)(bptr + k0);
        c = __builtin_amdgcn_wmma_f32_16x16x4_f32(false, a, false, bb,
                                                  (short)0, c, false, false);
    }
    float bv = bias[dcol];
    float* drow = dst + (rtile * 16 + 8 * half) * DD + dcol;
    #pragma unroll
    for (int r = 0; r < 8; ++r)
        drow[r * DD] = c[r] + bv;
}

// ---------------------------------------------------------------------------
// Kernel 2: aw = softmax_L(mask(q.k^T * scale)) ; f_baq = aw @ f_w ;
//           f_bq = f_b * (f_baq*lmask + f_s)
// One block of 256 threads per (b, ntile). Wave0 does WMMA + softmax -> LDS.
// ---------------------------------------------------------------------------
__global__ __launch_bounds__(256) void aw_kernel(
    const float* __restrict__ q, const float* __restrict__ kmat,
    const float* __restrict__ f_w, const float* __restrict__ f_b,
    const float* __restrict__ f_s, const float* __restrict__ query_mask,
    const float* __restrict__ lmask, float* __restrict__ f_bq)
{
    __shared__ float aw_s[16][LP];
    int b = blockIdx.x >> 3;
    int ntile = blockIdx.x & 7;
    int tid = threadIdx.x;

    if (tid < 32) {
        int half = tid >> 4, l16 = tid & 15;
        v8f c0 = {}, c1 = {};
        const float* aptr = q + (b * NN + ntile * 16 + l16) * DD + 2 * half;
        const float* b0p  = kmat + (b * LP + l16) * DD + 2 * half;
        const float* b1p  = kmat + (b * LP + 16 + l16) * DD + 2 * half;
        #pragma unroll 4
        for (int k0 = 0; k0 < DD; k0 += 4) {
            v2f a   = *(const v2f*)(aptr + k0);
            v2f bb0 = *(const v2f*)(b0p + k0);
            v2f bb1 = *(const v2f*)(b1p + k0);
            c0 = __builtin_amdgcn_wmma_f32_16x16x4_f32(false, a, false, bb0,
                                                       (short)0, c0, false, false);
            c1 = __builtin_amdgcn_wmma_f32_16x16x4_f32(false, a, false, bb1,
                                                       (short)0, c1, false, false);
        }
        float qm0 = query_mask[b * LL + l16];
        int col1 = 16 + l16;
        float qm1 = (col1 < LL) ? query_mask[b * LL + col1] : 0.0f;
        #pragma unroll
        for (int r = 0; r < 8; ++r) {
            float x0 = c0[r] * SCALE;
            x0 = (qm0 == 0.0f) ? NEGV : x0 * qm0;
            float x1 = NEGV;
            if (col1 < LL) {
                x1 = c1[r] * SCALE;
                x1 = (qm1 == 0.0f) ? NEGV : x1 * qm1;
            }
            float mx = fmaxf(x0, x1);
            for (int s = 8; s >= 1; s >>= 1)
                mx = fmaxf(mx, __shfl_xor(mx, s, 32));   // 16-lane group reduce
            float e0 = fast_exp(x0 - mx);
            float e1 = (col1 < LL) ? fast_exp(x1 - mx) : 0.0f;
            float sum = e0 + e1;
            for (int s = 8; s >= 1; s >>= 1)
                sum += __shfl_xor(sum, s, 32);
            float inv = __builtin_amdgcn_rcpf(sum);
            int rown = r + 8 * half;
            aw_s[rown][l16]      = e0 * inv;
            aw_s[rown][16 + l16] = e1 * inv;
        }
    }
    __syncthreads();

    int d = tid;                       // one d-column per thread
    float sv = f_s[b * DD + d];
    for (int r = 0; r < 16; ++r) {
        float acc = 0.0f;
        #pragma unroll 6
        for (int l = 0; l < LL; ++l)
            acc += aw_s[r][l] * f_w[(b * LL + l) * DD + d];
        int n = ntile * 16 + r;
        float fbm = lmask[b * NN + n];
        f_bq[(b * NN + n) * DD + d] =
            f_b[(b * NN + n) * DD + d] * (acc * fbm + sv);
    }
}

// ---------------------------------------------------------------------------
// Kernel 3: A = softmax_m(mask(f_bq.f_bq^T * scale)) * row_mask   (B,N,N)
// 256 threads per (b, ntile): 8 waves each WMMA one 16x16 m-tile -> LDS,
// then cooperative row softmax (16 threads/row, shuffle reduce).
// ---------------------------------------------------------------------------
__global__ __launch_bounds__(256) void attn_kernel(
    const float* __restrict__ f_bq, const float* __restrict__ lmask,
    float* __restrict__ A)
{
    __shared__ float As[16][NN];
    int b = blockIdx.x >> 3, ntile = blockIdx.x & 7;
    int tid = threadIdx.x;
    int wave = tid >> 5, lane = tid & 31;
    int half = lane >> 4, l16 = lane & 15;

    {
        v8f c = {};
        const float* aptr = f_bq + (b * NN + ntile * 16 + l16) * DD + 2 * half;
        const float* bptr = f_bq + (b * NN + wave * 16 + l16) * DD + 2 * half;
        #pragma unroll 4
        for (int k0 = 0; k0 < DD; k0 += 4) {
            v2f a  = *(const v2f*)(aptr + k0);
            v2f bb = *(const v2f*)(bptr + k0);
            c = __builtin_amdgcn_wmma_f32_16x16x4_f32(false, a, false, bb,
                                                      (short)0, c, false, false);
        }
        #pragma unroll
        for (int r = 0; r < 8; ++r)
            As[r + 8 * half][wave * 16 + l16] = c[r] * SCALE;
    }
    __syncthreads();

    int row = tid >> 4, cc = tid & 15;
    float v[8];
    float mx = -INFINITY;
    #pragma unroll
    for (int j = 0; j < 8; ++j) {
        int col = cc + 16 * j;
        float lm = lmask[b * NN + col];
        float x = As[row][col];
        x = (lm == 0.0f) ? NEGV : x * lm;
        v[j] = x;
        mx = fmaxf(mx, x);
    }
    for (int s = 8; s >= 1; s >>= 1)
        mx = fmaxf(mx, __shfl_xor(mx, s, 32));
    float sum = 0.0f;
    #pragma unroll
    for (int j = 0; j < 8; ++j) { v[j] = fast_exp(v[j] - mx); sum += v[j]; }
    for (int s = 8; s >= 1; s >>= 1)
        sum += __shfl_xor(sum, s, 32);
    float inv = __builtin_amdgcn_rcpf(sum);

    int n = ntile * 16 + row;
    float rowm = lmask[b * NN + n];
    #pragma unroll
    for (int j = 0; j < 8; ++j) {
        int col = cc + 16 * j;
        A[(b * NN + n) * NN + col] = v[j] * inv * rowm;
    }
}

// ---------------------------------------------------------------------------
// Kernel 4 (bandwidth-dominant): stream f_m (128 MB) exactly once.
// out[b,n,:] = A[b,n,:]@f_b[b]  +  f_b[b,n,:]  +  sum_m A[b,n,m]*sig(f_m*f_s)*f_m
// One 256-thread block per (b,n). f_m uses NON-TEMPORAL b128 loads so the
// one-shot 128 MB stream does not evict the L2-resident reused data
// (f_b: 1 MB, A: 0.5 MB, reused by 128 blocks per batch). global_prefetch
// issued 2 m-rows ahead to cover HBM latency.
// ---------------------------------------------------------------------------
__global__ __launch_bounds__(256) void fm_kernel(
    const float* __restrict__ A, const float* __restrict__ f_m,
    const float* __restrict__ f_b, const float* __restrict__ f_s,
    float* __restrict__ out)
{
    __shared__ float As[NN];
    __shared__ f4 red[256];
    __shared__ f4 redb[256];

    int b = blockIdx.x >> 7, n = blockIdx.x & 127;
    int tid = threadIdx.x;
    if (tid < NN) As[tid] = A[(b * NN + n) * NN + tid];
    __syncthreads();

    int grp = tid >> 6;            // m residue 0..3
    int d0  = (tid & 63) << 2;     // d base 0..252
    f4 s4 = *(const f4*)(f_s + b * DD + d0);
    f4 acc  = {0.f, 0.f, 0.f, 0.f};
    f4 accb = {0.f, 0.f, 0.f, 0.f};

    const float* fmBase = f_m + (size_t)(b * NN + n) * NN * DD;

    #pragma unroll 4
    for (int i = 0; i < 32; ++i) {
        int m = 4 * i + grp;
        const float* p = fmBase + (size_t)m * DD + d0;
        f4 fm = __builtin_nontemporal_load((const f4*)p);   // NT stream
        f4 fb = *(const f4*)(f_b + (b * NN + m) * DD + d0); // L2-resident
        float a = As[m];
        if (i + 2 < 32) __builtin_prefetch(p + 8 * DD, 0, 1);  // global_prefetch
        #pragma unroll
        for (int c = 0; c < 4; ++c) {
            float x = fm[c] * s4[c];
            float g = fast_sigmoid(x);
            acc[c]  += a * (g * fm[c]);
            accb[c] += a * fb[c];
        }
    }
    red[tid] = acc;
    redb[tid] = accb;
    __syncthreads();

    if (tid < 64) {
        f4 t = red[tid], tb = redb[tid];
        #pragma unroll
        for (int k = 1; k < 4; ++k) { t += red[tid + 64 * k]; tb += redb[tid + 64 * k]; }
        int dd = tid << 2;
        f4 fb0 = *(const f4*)(f_b + (b * NN + n) * DD + dd);
        f4 o = tb + fb0 + t;
        *(f4*)(out + (size_t)(b * NN + n) * DD + dd) = o;
    }
}

// ---------------------------------------------------------------------------
extern "C" void kernel_launch(void* const* d_in, const int* in_sizes, int n_in,
                              void* d_out, int out_size, void* d_ws, size_t ws_size,
                              hipStream_t stream) {
    const float* f_b   = (const float*)d_in[0];
    const float* f_w   = (const float*)d_in[1];
    const float* f_s   = (const float*)d_in[2];
    const float* f_m   = (const float*)d_in[3];
    const float* qmask = (const float*)d_in[4];
    const float* lmask = (const float*)d_in[5];
    const float* Wq    = (const float*)d_in[6];
    const float* bq    = (const float*)d_in[7];
    const float* Wk    = (const float*)d_in[8];
    const float* bk    = (const float*)d_in[9];
    float* outp = (float*)d_out;

    float* qws   = (float*)d_ws;               // B*N*D   = 262144 floats
    float* kws   = qws  + BB * NN * DD;        // B*LP*D  =  65536
    float* fbqws = kws  + BB * LP * DD;        // B*N*D   = 262144
    float* Aws   = fbqws + BB * NN * DD;       // B*N*N   = 131072

    qk_kernel<<<1280, 32, 0, stream>>>(f_b, f_w, Wq, bq, Wk, bk, qws, kws);
    aw_kernel<<<BB * 8, 256, 0, stream>>>(qws, kws, f_w, f_b, f_s, qmask, lmask, fbqws);
    attn_kernel<<<BB * 8, 256, 0, stream>>>(fbqws, lmask, Aws);
    fm_kernel<<<BB * NN, 256, 0, stream>>>(Aws, f_m, f_b, f_s, outp);
}